// MPNN_60198261621122
// MI455X (gfx1250) — compile-verified
//
#include <hip/hip_runtime.h>
#include <hip/hip_bf16.h>

// ---------------------------------------------------------------------------
// MPNN (REANN-style) forward + backward for MI455X / gfx1250 (wave32, WMMA).
// Dense contractions use V_WMMA_F32_16X16X4_F32 (exact f32, K=4):
//   - ctr = co[N,9,8] x cc[9,8,64]  (+ its backward through cc^T)
//   - MLP 64x64 layers (+ transposed-weight backward GEMMs)
// Edge scatter/gather passes are VALU + L2-resident global f32 atomics.
// ---------------------------------------------------------------------------

#define NW   8
#define NA   9
#define NCN  64
#define ROW  72                      // NA*NW
#define PIC  0.78539816339744831f    // pi / CUTOFF (CUTOFF = 4)

typedef __attribute__((ext_vector_type(2))) float v2f;
typedef __attribute__((ext_vector_type(8))) float v8f;

#define V8Z {0.f,0.f,0.f,0.f,0.f,0.f,0.f,0.f}

__device__ __forceinline__ v8f wmma4(v2f a, v2f b, v8f c) {
  // D = A(16x4 f32) * B(4x16 f32) + C(16x16 f32)
  return __builtin_amdgcn_wmma_f32_16x16x4_f32(false, a, false, b, (short)0, c,
                                               false, false);
}
__device__ __forceinline__ int il_of(int k){ return k==0 ? 0 : (k<4 ? 1 : 2); }

// ---------------------------------------------------------------------------
// Tiny species-embedding MLPs (din=1, dh=8): 4 species -> pure scalar code.
// ---------------------------------------------------------------------------
__device__ __forceinline__ void ln8_silu(float* h){
  float m=0.f;
  #pragma unroll
  for(int j=0;j<8;j++) m+=h[j];
  m*=0.125f;
  float v=0.f;
  #pragma unroll
  for(int j=0;j<8;j++){ float d=h[j]-m; v+=d*d; }
  v*=0.125f;
  float r=rsqrtf(v+1e-5f);
  #pragma unroll
  for(int j=0;j<8;j++){ float hh=(h[j]-m)*r; h[j]=hh/(1.f+expf(-hh)); }
}

__device__ __forceinline__ void emb_mlp(float x,
    const float* W1,const float* b1,const float* W2,const float* b2,
    const float* W3,const float* b3, float* out, int dout){
  float h1[8],h2[8];
  #pragma unroll
  for(int j=0;j<8;j++) h1[j]=x*W1[j]+b1[j];
  ln8_silu(h1);
  for(int j=0;j<8;j++){ float s=b2[j]; for(int k=0;k<8;k++) s+=h1[k]*W2[k*8+j]; h2[j]=s; }
  ln8_silu(h2);
  for(int o=0;o<dout;o++){ float s=b3[o]; for(int k=0;k<8;k++) s+=h2[k]*W3[k*dout+o]; out[o]=s; }
}

__global__ void k_emb(const float* __restrict__ aspec,
    const float* cW1,const float* cb1,const float* cW2,const float* cb2,
    const float* cW3,const float* cb3,
    const float* nW1,const float* nb1,const float* nW2,const float* nb2,
    const float* nW3,const float* nb3,
    float* __restrict__ embc, float* __restrict__ embn, int T){
  int s=threadIdx.x; if (s>=T) return;
  float x=aspec[s];
  emb_mlp(x,cW1,cb1,cW2,cb2,cW3,cb3, embc+s*64, 64);
  emb_mlp(x,nW1,nb1,nW2,nb2,nW3,nb3, embn+s*24, 24);
}

// ---------------------------------------------------------------------------
// Solid harmonics l<=2
// ---------------------------------------------------------------------------
#define SC0  0.28209479177387814f
#define SC1  0.4886025119029199f
#define SC2A 1.0925484305920792f
#define SC2B 0.31539156525252005f
#define SC2C 0.5462742152960396f

__device__ __forceinline__ void sph_fill(float dx,float dy,float dz,float r2,float* s){
  s[0]=SC0; s[1]=SC1*dy; s[2]=SC1*dz; s[3]=SC1*dx;
  s[4]=SC2A*dx*dy; s[5]=SC2A*dy*dz; s[6]=SC2B*(3.f*dz*dz-r2);
  s[7]=SC2A*dx*dz; s[8]=SC2C*(dx*dx-dy*dy);
}

// ---------------------------------------------------------------------------
// Edge forward: scatter orbital (t==0) or wo (t>=1) into co_out.
// All operands (cart, ic, co_prev, co_out) are L2-resident.
// ---------------------------------------------------------------------------
__global__ void k_edge_fwd(const float* __restrict__ cart,
    const int* __restrict__ cl, const int* __restrict__ nl,
    const int* __restrict__ nsp, const int* __restrict__ cnl,
    const float* __restrict__ embn,
    const float* __restrict__ ic,        // null for t==0
    const float* __restrict__ co_prev,   // null for t==0
    float* __restrict__ co_out, int E){
  int e = blockIdx.x*blockDim.x + threadIdx.x;
  if (e>=E) return;
  int c=cl[e], n=nl[e], s=nsp[e];
  float dx=cart[3*c]-cart[3*n], dy=cart[3*c+1]-cart[3*n+1], dz=cart[3*c+2]-cart[3*n+2];
  float r2=dx*dx+dy*dy+dz*dz;
  float dist=sqrtf(r2);
  float q=0.5f*cosf(dist*PIC)+0.5f;
  float cut=q*q;
  float sph[9]; sph_fill(dx,dy,dz,r2,sph);
  const float* nb = embn + s*24;
  float P[NW];
  #pragma unroll
  for(int w=0;w<NW;w++){
    float t0=dist-nb[16+w], a=nb[8+w];
    P[w]=cut*expf(-a*a*t0*t0)*nb[w];
  }
  if (!ic){
    for(int k=0;k<NA;k++)
      #pragma unroll
      for(int w=0;w<NW;w++)
        atomicAdd(&co_out[(size_t)c*ROW+k*NW+w], P[w]*sph[k]);
  } else {
    int cn=cnl[e];
    float icv[NW];
    #pragma unroll
    for(int w=0;w<NW;w++) icv[w]=ic[(size_t)cn*NW+w];
    const float* cp = co_prev + (size_t)cn*ROW;
    for(int k=0;k<NA;k++)
      #pragma unroll
      for(int w=0;w<NW;w++){
        float wo = icv[w]*P[w]*sph[k] + cp[k*NW+w]*cut;
        atomicAdd(&co_out[(size_t)c*ROW+k*NW+w], wo);
      }
  }
}

// ---------------------------------------------------------------------------
// Edge backward. iter mode (ic!=null): wo = ic[cn]*orb + co_prev[cn]*cut.
// t==0 mode (ic==null): co_0 = scatter(orbital) -> g_orb = g_co[c].
// ---------------------------------------------------------------------------
__global__ void k_edge_bwd(const float* __restrict__ cart,
    const int* __restrict__ cl, const int* __restrict__ nl,
    const int* __restrict__ nsp, const int* __restrict__ cnl,
    const float* __restrict__ embn,
    const float* __restrict__ gcoR,      // level-t grad (read)
    float* __restrict__ gcoW,            // level-(t-1) grad (accumulate), null at t==0
    const float* __restrict__ co_prev,   // null at t==0
    const float* __restrict__ ic,        // null at t==0
    float* __restrict__ g_ic,            // null at t==0
    float* __restrict__ g_cart, int E){
  int e = blockIdx.x*blockDim.x + threadIdx.x;
  if (e>=E) return;
  int c=cl[e], n=nl[e], s=nsp[e];
  float dx=cart[3*c]-cart[3*n], dy=cart[3*c+1]-cart[3*n+1], dz=cart[3*c+2]-cart[3*n+2];
  float r2=dx*dx+dy*dy+dz*dz;
  float dist=sqrtf(r2);
  float q=0.5f*cosf(dist*PIC)+0.5f;
  float cut=q*q;
  float sph[9]; sph_fill(dx,dy,dz,r2,sph);
  const float* nb = embn + s*24;
  float rad[NW], amp[NW], alp[NW], dmu[NW];
  #pragma unroll
  for(int w=0;w<NW;w++){
    amp[w]=nb[w]; alp[w]=nb[8+w]; dmu[w]=dist-nb[16+w];
    rad[w]=expf(-alp[w]*alp[w]*dmu[w]*dmu[w]);
  }
  bool iter = (ic!=nullptr);
  int cn = iter ? cnl[e] : 0;
  float icv[NW]; float gic[NW];
  #pragma unroll
  for(int w=0;w<NW;w++){ icv[w]= iter ? ic[(size_t)cn*NW+w] : 0.f; gic[w]=0.f; }
  float gcut=0.f, gsph[NA], gradw[NW];
  #pragma unroll
  for(int k=0;k<NA;k++) gsph[k]=0.f;
  #pragma unroll
  for(int w=0;w<NW;w++) gradw[w]=0.f;

  for(int k=0;k<NA;k++){
    #pragma unroll
    for(int w=0;w<NW;w++){
      float gw = gcoR[(size_t)c*ROW+k*NW+w];
      float gorb;
      if (iter){
        float cp = co_prev[(size_t)cn*ROW+k*NW+w];
        atomicAdd(&gcoW[(size_t)cn*ROW+k*NW+w], gw*cut);
        gic[w] += gw * (cut*rad[w]*amp[w]*sph[k]);
        gcut   += gw * cp;
        gorb = gw * icv[w];
      } else {
        gorb = gw;
      }
      float raw = rad[w]*amp[w];
      gcut     += gorb*raw*sph[k];
      gradw[w] += gorb*cut*amp[w]*sph[k];
      gsph[k]  += gorb*cut*raw;
    }
  }
  if (iter){
    #pragma unroll
    for(int w=0;w<NW;w++) atomicAdd(&g_ic[(size_t)cn*NW+w], gic[w]);
  }
  // d(cut)/d(dist) = 2q * (-0.5*PIC*sin(PIC*dist))
  float gdist = gcut * (2.f*q) * (-0.5f*PIC*sinf(dist*PIC));
  #pragma unroll
  for(int w=0;w<NW;w++)
    gdist += gradw[w] * rad[w] * (-2.f*alp[w]*alp[w]*dmu[w]);
  float invd = 1.f/dist;   // matches reference NaN semantics at dist==0
  float gx=gdist*dx*invd, gy=gdist*dy*invd, gz=gdist*dz*invd;
  gx += gsph[3]*SC1 + gsph[4]*SC2A*dy + gsph[6]*(-2.f*SC2B*dx) + gsph[7]*SC2A*dz + gsph[8]*2.f*SC2C*dx;
  gy += gsph[1]*SC1 + gsph[4]*SC2A*dx + gsph[5]*SC2A*dz + gsph[6]*(-2.f*SC2B*dy) + gsph[8]*(-2.f*SC2C*dy);
  gz += gsph[2]*SC1 + gsph[5]*SC2A*dy + gsph[6]*(4.f*SC2B*dz) + gsph[7]*SC2A*dx;
  atomicAdd(&g_cart[3*c+0], gx); atomicAdd(&g_cart[3*c+1], gy); atomicAdd(&g_cart[3*c+2], gz);
  atomicAdd(&g_cart[3*n+0],-gx); atomicAdd(&g_cart[3*n+1],-gy); atomicAdd(&g_cart[3*n+2],-gz);
}

// ---------------------------------------------------------------------------
// Contract: ctr[i,k,m] = sum_j co[i,k,j]*cc[IL(k),j,m]   (WMMA f32 16x16x4)
// dens_out[i,m] = dens_prev[i,m] + lc[i,m] * sum_k ctr^2
// One wave per 16-atom tile, 2 waves/block. cc tile (18KB) in LDS.
// ---------------------------------------------------------------------------
__global__ __launch_bounds__(64) void k_contract(
    const float* __restrict__ co, const float* __restrict__ cc_t,
    const float* __restrict__ embc, const int* __restrict__ lspec,
    const float* __restrict__ dens_prev, float* __restrict__ dens_out, int N){
  __shared__ float ccL[NA*512];
  for (int i=threadIdx.x;i<NA*512;i+=64) ccL[i]=cc_t[il_of(i>>9)*512+(i&511)];
  __syncthreads();
  int wv=threadIdx.x>>5, lane=threadIdx.x&31, hb=lane>>4, l16=lane&15;
  int atom0=blockIdx.x*32+wv*16;
  if (atom0>=N) return;
  v8f dacc[4]={V8Z,V8Z,V8Z,V8Z};
  const float* coR = co + (size_t)(atom0+l16)*ROW;
  for (int k=0;k<NA;k++){
    const float* A=coR+k*NW;
    float a0=A[2*hb],a1=A[2*hb+1],a2=A[4+2*hb],a3=A[5+2*hb];
    const float* B=ccL+k*512;
    #pragma unroll
    for (int mt=0;mt<4;mt++){
      v8f acc=V8Z; v2f av,bv;
      av.x=a0; av.y=a1;
      bv.x=B[(2*hb)*64+mt*16+l16]; bv.y=B[(2*hb+1)*64+mt*16+l16];
      acc=wmma4(av,bv,acc);
      av.x=a2; av.y=a3;
      bv.x=B[(4+2*hb)*64+mt*16+l16]; bv.y=B[(5+2*hb)*64+mt*16+l16];
      acc=wmma4(av,bv,acc);
      #pragma unroll
      for (int r=0;r<8;r++) dacc[mt][r]+=acc[r]*acc[r];
    }
  }
  #pragma unroll
  for (int mt=0;mt<4;mt++){
    #pragma unroll
    for (int r=0;r<8;r++){
      int i=atom0+r+8*hb, m=mt*16+l16;
      float v=dacc[mt][r]*embc[lspec[i]*64+m];
      if (dens_prev) v+=dens_prev[(size_t)i*64+m];
      dens_out[(size_t)i*64+m]=v;
    }
  }
}

// ---------------------------------------------------------------------------
// Contract backward: g_ctr = 2*ctr*lc*g_dens (recompute ctr via WMMA),
// then g_co[:,k,:] += g_ctr @ cc[k]^T  (K=64 WMMA, N=8 padded to 16).
// ---------------------------------------------------------------------------
__global__ __launch_bounds__(64) void k_contract_bwd(
    const float* __restrict__ co, const float* __restrict__ cc_t,
    const float* __restrict__ embc, const int* __restrict__ lspec,
    const float* __restrict__ gD, float* __restrict__ gCo, int N){
  __shared__ float ccL[NA*512];
  __shared__ float gctr[2][16*64];
  for (int i=threadIdx.x;i<NA*512;i+=64) ccL[i]=cc_t[il_of(i>>9)*512+(i&511)];
  __syncthreads();
  int wv=threadIdx.x>>5, lane=threadIdx.x&31, hb=lane>>4, l16=lane&15;
  int atom0=blockIdx.x*32+wv*16;
  if (atom0>=N) return;
  float* G=gctr[wv];
  const float* coR=co+(size_t)(atom0+l16)*ROW;
  for (int k=0;k<NA;k++){
    const float* A=coR+k*NW;
    float a0=A[2*hb],a1=A[2*hb+1],a2=A[4+2*hb],a3=A[5+2*hb];
    const float* B=ccL+k*512;
    #pragma unroll
    for (int mt=0;mt<4;mt++){
      v8f acc=V8Z; v2f av,bv;
      av.x=a0; av.y=a1;
      bv.x=B[(2*hb)*64+mt*16+l16]; bv.y=B[(2*hb+1)*64+mt*16+l16];
      acc=wmma4(av,bv,acc);
      av.x=a2; av.y=a3;
      bv.x=B[(4+2*hb)*64+mt*16+l16]; bv.y=B[(5+2*hb)*64+mt*16+l16];
      acc=wmma4(av,bv,acc);
      #pragma unroll
      for (int r=0;r<8;r++){
        int row=r+8*hb, i=atom0+row, m=mt*16+l16;
        G[row*64+m]=2.f*acc[r]*embc[lspec[i]*64+m]*gD[(size_t)i*64+m];
      }
    }
    // LDS ops are in-order within a wave -> G visible to A-layout reads below.
    v8f gacc=V8Z;
    for (int kk=0;kk<64;kk+=4){
      v2f av,bv;
      av.x=G[l16*64+kk+2*hb]; av.y=G[l16*64+kk+2*hb+1];
      bv.x=(l16<NW)?B[l16*64+kk+2*hb]:0.f;     // B^T[kr][j] = cc[k][j][kr]
      bv.y=(l16<NW)?B[l16*64+kk+2*hb+1]:0.f;
      gacc=wmma4(av,bv,gacc);
    }
    if (l16<NW){
      #pragma unroll
      for (int r=0;r<8;r++){
        int i=atom0+r+8*hb;
        gCo[(size_t)i*ROW+k*NW+l16]+=gacc[r];
      }
    }
  }
}

// ---------------------------------------------------------------------------
// 64->64 WMMA layer helpers (activations in LDS, weights streamed from L2)
// ---------------------------------------------------------------------------
__device__ __forceinline__ void layer64(const float* __restrict__ W,
    const float* __restrict__ b, const float* actIn, float* actOut,
    int hb, int l16){
  #pragma unroll
  for (int mt=0;mt<4;mt++){
    v8f acc=V8Z;
    for (int kk=0;kk<64;kk+=4){
      v2f av,bv;
      av.x=actIn[l16*64+kk+2*hb];       av.y=actIn[l16*64+kk+2*hb+1];
      bv.x=W[(kk+2*hb)*64+mt*16+l16];   bv.y=W[(kk+2*hb+1)*64+mt*16+l16];
      acc=wmma4(av,bv,acc);
    }
    float bb = b ? b[mt*16+l16] : 0.f;
    #pragma unroll
    for (int r=0;r<8;r++) actOut[(r+8*hb)*64+mt*16+l16]=acc[r]+bb;
  }
}
__device__ __forceinline__ void layer64T(const float* __restrict__ W,
    const float* gIn, float* gOut, int hb, int l16){
  #pragma unroll
  for (int mt=0;mt<4;mt++){
    v8f acc=V8Z;
    for (int kk=0;kk<64;kk+=4){
      v2f av,bv;
      av.x=gIn[l16*64+kk+2*hb];           av.y=gIn[l16*64+kk+2*hb+1];
      bv.x=W[(mt*16+l16)*64+kk+2*hb];     bv.y=W[(mt*16+l16)*64+kk+2*hb+1];
      acc=wmma4(av,bv,acc);
    }
    #pragma unroll
    for (int r=0;r<8;r++) gOut[(r+8*hb)*64+mt*16+l16]=acc[r];
  }
}
__device__ __forceinline__ void ln_stats(const float* act, float* st, int lane){
  if (lane<16){
    float s=0.f,s2=0.f;
    for (int j=0;j<64;j++){ float v=act[lane*64+j]; s+=v; s2+=v*v; }
    float m=s*(1.f/64.f), var=s2*(1.f/64.f)-m*m;
    st[lane*2]=m; st[lane*2+1]=rsqrtf(var+1e-5f);
  }
}
__device__ __forceinline__ void ln_silu_apply(float* act, const float* st, int lane){
  for (int q2=lane;q2<1024;q2+=32){
    int row=q2>>6;
    float h=(act[q2]-st[row*2])*st[row*2+1];
    act[q2]=h/(1.f+expf(-h));
  }
}
// z -> h (kept, in place), a = silu(h), rstd saved
__device__ __forceinline__ void ln_silu_keep(float* z_h, float* a, float* st,
                                             float* rstd, int lane){
  ln_stats(z_h,st,lane);
  if (lane<16) rstd[lane]=st[lane*2+1];
  for (int q2=lane;q2<1024;q2+=32){
    int row=q2>>6;
    float h=(z_h[q2]-st[row*2])*st[row*2+1];
    z_h[q2]=h;
    a[q2]=h/(1.f+expf(-h));
  }
}
// gA (wrt silu out) + h + rstd -> gZ (wrt pre-LN). gA may alias gZ.
__device__ __forceinline__ void silu_ln_bwd(const float* gA, const float* h,
    const float* rstd, float* gZ, float* st, int lane){
  for (int q2=lane;q2<1024;q2+=32){
    float hv=h[q2], s=1.f/(1.f+expf(-hv));
    gZ[q2]=gA[q2]*s*(1.f+hv*(1.f-s));
  }
  if (lane<16){
    float s1=0.f,s2=0.f;
    for (int j=0;j<64;j++){ float g=gZ[lane*64+j]; s1+=g; s2+=g*h[lane*64+j]; }
    st[lane*2]=s1*(1.f/64.f); st[lane*2+1]=s2*(1.f/64.f);
  }
  for (int q2=lane;q2<1024;q2+=32){
    int row=q2>>6;
    gZ[q2]=rstd[row]*(gZ[q2]-st[row*2]-h[q2]*st[row*2+1]);
  }
}

// ---------------------------------------------------------------------------
// MLP forward (dh=64). dout<=8 final layer in VALU.
// outI != null -> write ic[N,dout];  outE != null -> write output + energy.
// ---------------------------------------------------------------------------
__global__ __launch_bounds__(64) void k_mlp_fwd(
    const float* __restrict__ x,
    const float* W1,const float* b1,const float* W2,const float* b2,
    const float* W3,const float* b3,
    float* __restrict__ outI, float* __restrict__ outE, float* __restrict__ energy,
    int dout, int N){
  __shared__ float buf[2][2*1024+32];
  int wv=threadIdx.x>>5, lane=threadIdx.x&31, hb=lane>>4, l16=lane&15;
  int atom0=blockIdx.x*32+wv*16;
  if (atom0>=N) return;
  float* B0=buf[wv]; float* B1=B0+1024; float* st=B1+1024;
  for (int q2=lane;q2<1024;q2+=32)
    B0[q2]=x[(size_t)(atom0+(q2>>6))*64+(q2&63)];
  layer64(W1,b1,B0,B1,hb,l16);
  ln_stats(B1,st,lane); ln_silu_apply(B1,st,lane);
  layer64(W2,b2,B1,B0,hb,l16);
  ln_stats(B0,st,lane); ln_silu_apply(B0,st,lane);
  for (int idx=lane; idx<16*dout; idx+=32){
    int row=idx/dout, c=idx%dout;
    float s=b3[c];
    for (int j=0;j<64;j++) s+=B0[row*64+j]*W3[j*dout+c];
    int i=atom0+row;
    if (outI) outI[(size_t)i*dout+c]=s;
    if (outE){ outE[i]=s; atomicAdd(energy,s); }
  }
}

// ---------------------------------------------------------------------------
// MLP backward: recompute fwd, then gX = J^T gOut (gOut==null -> all ones).
// ---------------------------------------------------------------------------
__global__ __launch_bounds__(64) void k_mlp_bwd(
    const float* __restrict__ x,
    const float* W1,const float* b1,const float* W2,const float* b2,const float* W3,
    const float* __restrict__ gOut,
    float* __restrict__ gX, int accumulate, int dout, int N){
  __shared__ float buf[2][4*1024+32+16+16];
  int wv=threadIdx.x>>5, lane=threadIdx.x&31, hb=lane>>4, l16=lane&15;
  int atom0=blockIdx.x*32+wv*16;
  if (atom0>=N) return;
  float* B0=buf[wv]; float* B1=B0+1024; float* B2=B1+1024; float* B3=B2+1024;
  float* st=B3+1024; float* r1=st+32; float* r2=r1+16;
  for (int q2=lane;q2<1024;q2+=32)
    B0[q2]=x[(size_t)(atom0+(q2>>6))*64+(q2&63)];
  layer64(W1,b1,B0,B1,hb,l16);          // z1 in B1
  ln_silu_keep(B1,B2,st,r1,lane);       // h1 in B1, a1 in B2
  layer64(W2,b2,B2,B3,hb,l16);          // z2 in B3
  ln_silu_keep(B3,B2,st,r2,lane);       // h2 in B3, a2 in B2 (unused further)
  // g_a2 = gOut @ W3^T  -> B0 (x dead)
  for (int q2=lane;q2<1024;q2+=32){
    int row=q2>>6, j=q2&63, i=atom0+row;
    float s=0.f;
    for (int c=0;c<dout;c++){
      float g = gOut ? gOut[(size_t)i*dout+c] : 1.f;
      s += g*W3[j*dout+c];
    }
    B0[q2]=s;
  }
  silu_ln_bwd(B0,B3,r2,B0,st,lane);     // g_z2 in B0
  layer64T(W2,B0,B2,hb,l16);            // g_a1 in B2
  silu_ln_bwd(B2,B1,r1,B2,st,lane);     // g_z1 in B2
  layer64T(W1,B2,B3,hb,l16);            // g_x in B3
  for (int q2=lane;q2<1024;q2+=32){
    size_t idx=(size_t)(atom0+(q2>>6))*64+(q2&63);
    if (accumulate) gX[idx]+=B3[q2]; else gX[idx]=B3[q2];
  }
}

__global__ void k_finalize(const float* __restrict__ gC, float* __restrict__ og, int n3){
  int i=blockIdx.x*blockDim.x+threadIdx.x;
  if (i<n3) og[i]=-gC[i];
}

// ---------------------------------------------------------------------------
// Host orchestration
// ---------------------------------------------------------------------------
extern "C" void kernel_launch(void* const* d_in, const int* in_sizes, int n_in,
                              void* d_out, int out_size, void* d_ws, size_t ws_size,
                              hipStream_t stream)
{
  (void)out_size; (void)ws_size;
  auto F=[&](int i){ return (const float*)d_in[i]; };
  auto I=[&](int i){ return (const int*)d_in[i]; };

  // Resolve harness flattening order (insertion-order vs fully-sorted tree).
  int iCart,iAs,iCl,iNl,iLs,iNs,iCnl,pB;
  if (in_sizes[0] > in_sizes[1]) {       // cart(3N) first -> insertion order
    iCart=0; iAs=1; pB=2;
    iCl=n_in-5; iNl=n_in-4; iLs=n_in-3; iNs=n_in-2; iCnl=n_in-1;
  } else {                               // atom_species(4) first -> sorted
    iAs=0; iCart=1; iCnl=2; iCl=3; iLs=4; iNs=5; iNl=6; pB=7;
  }
  const int N=in_sizes[iLs], E=in_sizes[iCl];

  struct MLPP { const float *W1,*b1,*W2,*b2,*W3,*b3; };
  bool ccFirst = (in_sizes[pB] == 4*3*NW*NCN);        // contracted_coeff = 6144
  int m0 = ccFirst ? pB+1 : pB;
  bool sortedF = (in_sizes[m0+1] == 64);              // W2(64) vs b1(8)
  auto getM=[&](int b)->MLPP{
    MLPP m;
    if (sortedF){ m.W1=F(b); m.W2=F(b+1); m.W3=F(b+2); m.b1=F(b+3); m.b2=F(b+4); m.b3=F(b+5); }
    else        { m.W1=F(b); m.b1=F(b+1); m.W2=F(b+2); m.b2=F(b+3); m.W3=F(b+4); m.b3=F(b+5); }
    return m;
  };
  const float* ccp; MLPP embC,embN,it[3],onn;
  if (ccFirst){
    ccp=F(pB); embC=getM(pB+1); embN=getM(pB+7);
    it[0]=getM(pB+13); it[1]=getM(pB+19); it[2]=getM(pB+25); onn=getM(pB+31);
  } else {
    embN=getM(pB); embC=getM(pB+6);
    it[0]=getM(pB+12); it[1]=getM(pB+18); it[2]=getM(pB+24);
    onn=getM(pB+30); ccp=F(pB+36);
  }

  // Workspace layout (floats) -- everything hot stays L2-resident.
  float* ws=(float*)d_ws;
  float* embc = ws;                          // 4*64
  float* embn = embc + 4*64;                 // 4*24
  float* coB  = embn + 4*24;                 // 4 * N*72
  float* dnsB = coB  + (size_t)4*N*ROW;      // 4 * N*64
  float* icB  = dnsB + (size_t)4*N*NCN;      // 3 * N*8
  float* gD   = icB  + (size_t)3*N*NW;       // N*64
  float* gA   = gD   + (size_t)N*NCN;        // N*72
  float* gB   = gA   + (size_t)N*ROW;        // N*72
  float* gI   = gB   + (size_t)N*ROW;        // N*8
  float* gC   = gI   + (size_t)N*NW;         // N*3
  auto CO=[&](int t){ return coB  + (size_t)t*N*ROW; };
  auto DN=[&](int t){ return dnsB + (size_t)t*N*NCN; };
  auto IC=[&](int t){ return icB  + (size_t)(t-1)*N*NW; };

  float* outp=(float*)d_out;
  float* energy=outp; float* ograd=outp+1; float* oout=outp+1+(size_t)3*N;

  const float* cart=F(iCart); const float* aspec=F(iAs);
  const int *cl=I(iCl),*nl=I(iNl),*ls=I(iLs),*nsp=I(iNs),*cnl=I(iCnl);

  const int EB=256, EG=(E+EB-1)/EB, CG=(N+31)/32;

  // ---- forward ----
  k_emb<<<1,32,0,stream>>>(aspec,
      embC.W1,embC.b1,embC.W2,embC.b2,embC.W3,embC.b3,
      embN.W1,embN.b1,embN.W2,embN.b2,embN.W3,embN.b3, embc,embn,4);
  hipMemsetAsync(CO(0),0,(size_t)N*ROW*sizeof(float),stream);
  k_edge_fwd<<<EG,EB,0,stream>>>(cart,cl,nl,nsp,cnl,embn,nullptr,nullptr,CO(0),E);
  k_contract<<<CG,64,0,stream>>>(CO(0),ccp,embc,ls,nullptr,DN(0),N);
  for (int t=1;t<=3;t++){
    MLPP& mp=it[t-1];
    k_mlp_fwd<<<CG,64,0,stream>>>(DN(t-1),mp.W1,mp.b1,mp.W2,mp.b2,mp.W3,mp.b3,
                                  IC(t),nullptr,nullptr,8,N);
    hipMemcpyAsync(CO(t),CO(t-1),(size_t)N*ROW*sizeof(float),
                   hipMemcpyDeviceToDevice,stream);
    k_edge_fwd<<<EG,EB,0,stream>>>(cart,cl,nl,nsp,cnl,embn,IC(t),CO(t-1),CO(t),E);
    k_contract<<<CG,64,0,stream>>>(CO(t),ccp+(size_t)t*3*NW*NCN,embc,ls,
                                   DN(t-1),DN(t),N);
  }
  hipMemsetAsync(energy,0,sizeof(float),stream);
  k_mlp_fwd<<<CG,64,0,stream>>>(DN(3),onn.W1,onn.b1,onn.W2,onn.b2,onn.W3,onn.b3,
                                nullptr,oout,energy,1,N);

  // ---- backward (grad wrt cart of sum(output)) ----
  k_mlp_bwd<<<CG,64,0,stream>>>(DN(3),onn.W1,onn.b1,onn.W2,onn.b2,onn.W3,
                                nullptr,gD,0,1,N);      // gD = dE/d density_3
  hipMemsetAsync(gA,0,(size_t)N*ROW*sizeof(float),stream);
  hipMemsetAsync(gC,0,(size_t)N*3*sizeof(float),stream);
  float* gcur=gA; float* gnxt=gB;
  for (int t=3;t>=1;t--){
    MLPP& mp=it[t-1];
    k_contract_bwd<<<CG,64,0,stream>>>(CO(t),ccp+(size_t)t*3*NW*NCN,embc,ls,
                                       gD,gcur,N);       // density_t term
    hipMemcpyAsync(gnxt,gcur,(size_t)N*ROW*sizeof(float),
                   hipMemcpyDeviceToDevice,stream);      // identity co path
    hipMemsetAsync(gI,0,(size_t)N*NW*sizeof(float),stream);
    k_edge_bwd<<<EG,EB,0,stream>>>(cart,cl,nl,nsp,cnl,embn,
                                   gcur,gnxt,CO(t-1),IC(t),gI,gC,E);
    k_mlp_bwd<<<CG,64,0,stream>>>(DN(t-1),mp.W1,mp.b1,mp.W2,mp.b2,mp.W3,
                                  gI,gD,1,8,N);          // gD -> dE/d density_{t-1}
    float* tmp=gcur; gcur=gnxt; gnxt=tmp;
  }
  k_contract_bwd<<<CG,64,0,stream>>>(CO(0),ccp,embc,ls,gD,gcur,N);
  k_edge_bwd<<<EG,EB,0,stream>>>(cart,cl,nl,nsp,cnl,embn,
                                 gcur,nullptr,nullptr,nullptr,nullptr,gC,E);
  k_finalize<<<((3*N)+255)/256,256,0,stream>>>(gC,ograd,3*N);
}